// MyDetector_59304908423417
// MI455X (gfx1250) — compile-verified
//
#include <hip/hip_runtime.h>
#include <math.h>

#define CLASS_NUM 80
#define NBOX      10647          // 507 + 2028 + 8112
#define NPAD      10656          // multiple of 32, row stride for scores (16B-aligned * 4B)
#define KNMS      100
#define BATCH     32
#define IOU_T     0.3f

__device__ __forceinline__ float sigmoidf_(float x) { return 1.0f / (1.0f + __expf(-x)); }

// ---------------------------------------------------------------------------
// Decode one YOLO scale. Thread = (b, anchor, y, x) with x innermost so every
// channel read is a fully coalesced 32-lane contiguous load (116 MB streamed
// once -> HBM-bound, ~5us at 23.3 TB/s). S is a template parameter so all
// idx%S / idx/S decompositions strength-reduce to mul/shift (no v_s_rcp_f32
// division preamble).
// ---------------------------------------------------------------------------
template <int S>
__global__ void decode_kernel(const float* __restrict__ p,
                              const float* __restrict__ anc,      // 6 floats (3 anchors x wh)
                              const float* __restrict__ thresh_p,
                              float stride, int boxBase,
                              float* __restrict__ boxes,          // [B, NBOX, 6]
                              float* __restrict__ scores)         // [B, NPAD]
{
    constexpr int SS    = S * S;
    constexpr int total = BATCH * 3 * SS;
    int idx = blockIdx.x * blockDim.x + threadIdx.x;
    if (idx >= total) return;

    int x = idx % S;
    int t = idx / S;
    int y = t % S;  t /= S;
    int a = t % 3;
    int b = t / 3;

    size_t base = ((size_t)b * 255 + (size_t)a * 85) * (size_t)SS + (size_t)y * S + x;
    const float thresh = *thresh_p;

    float conf = sigmoidf_(p[base]);
    float tx   = sigmoidf_(p[base + (size_t)SS]);
    float ty   = sigmoidf_(p[base + 2 * (size_t)SS]);
    float w    = __expf(p[base + 3 * (size_t)SS]) * anc[a * 2 + 0];
    float h    = __expf(p[base + 4 * (size_t)SS]) * anc[a * 2 + 1];
    float ox   = ((float)x + tx) * stride;
    float oy   = ((float)y + ty) * stride;

    // argmax over 80 class logits (first-max wins, matching jnp.argmax).
    // Unroll so multiple independent global loads are in flight per wave.
    float best = p[base + 5 * (size_t)SS];
    int bi = 0;
    #pragma unroll 8
    for (int c = 1; c < CLASS_NUM; ++c) {
        float v = p[base + (size_t)(5 + c) * SS];
        if (v > best) { best = v; bi = c; }
    }

    int n = (y * S + x) * 3 + a + boxBase;
    float* o = boxes + ((size_t)b * NBOX + n) * 6;
    o[0] = ox - 0.5f * w;
    o[1] = oy - 0.5f * h;
    o[2] = ox + 0.5f * w;
    o[3] = oy + 0.5f * h;
    o[4] = (float)bi;
    o[5] = conf;

    scores[(size_t)b * NPAD + n] = (conf > thresh) ? conf : 0.0f;
}

// ---------------------------------------------------------------------------
// Per-image top-100 (exact sorted order, tie -> lower index) + greedy NMS.
// One block per image, 1024 threads (32 wave32s).
// Score row is staged into LDS with the CDNA5 async data mover:
//   global_load_async_to_lds_b128 (GVS: SGPR base + per-thread u32 offset)
//   + s_wait_asynccnt 0.
// ---------------------------------------------------------------------------
__global__ void __launch_bounds__(1024)
topk_nms_kernel(const float* __restrict__ boxes,
                const float* __restrict__ scores,
                const float* __restrict__ thresh_p,
                float* __restrict__ out_sel,     // [B, 100, 6]
                float* __restrict__ out_keep)    // [B, 100]
{
    __shared__ __attribute__((aligned(16))) float s_scores[NPAD];  // 42.6 KB
    __shared__ unsigned            s_selidx[KNMS];
    __shared__ float               s_selbox[KNMS * 6];
    __shared__ int                 s_keep[KNMS];
    __shared__ unsigned long long  s_wmax[32];

    const int tid  = threadIdx.x;
    const int wave = tid >> 5;
    const int lane = tid & 31;
    const int b    = blockIdx.x;

    // ---- async copy scores row (NPAD floats = 2664 x b128) global -> LDS ----
    const float* row = scores + (size_t)b * NPAD;   // uniform => SGPR base (GVS mode)
    unsigned lds0 = (unsigned)(size_t)(void*)s_scores;  // low 32 bits = LDS offset
    for (int c = tid; c < NPAD / 4; c += 1024) {
        unsigned goff  = (unsigned)c * 16u;
        unsigned laddr = lds0 + (unsigned)c * 16u;
        asm volatile("global_load_async_to_lds_b128 %0, %1, %2"
                     :: "v"(laddr), "v"(goff), "s"(row) : "memory");
    }
    asm volatile("s_wait_asynccnt 0" ::: "memory");
    __syncthreads();
    // padding slots -> "removed" sentinel (negative => key 0, never selected)
    for (int j = NBOX + tid; j < NPAD; j += 1024) s_scores[j] = -1.0f;
    __syncthreads();

    // ---- 100 rounds of argmax with packed (score_bits, ~idx) u64 keys ----
    for (int k = 0; k < KNMS; ++k) {
        unsigned long long best = 0ull;
        for (int j = tid; j < NPAD; j += 1024) {
            unsigned sb = __float_as_uint(s_scores[j]);
            unsigned long long key = (sb & 0x80000000u)
                ? 0ull
                : (((unsigned long long)sb << 32) |
                   (unsigned long long)(0xFFFFFFFFu - (unsigned)j));
            if (key > best) best = key;
        }
        // wave32 reduction
        for (int off = 16; off > 0; off >>= 1) {
            unsigned lo = (unsigned)best, hi = (unsigned)(best >> 32);
            unsigned olo = __shfl_xor(lo, off, 32);
            unsigned ohi = __shfl_xor(hi, off, 32);
            unsigned long long o = ((unsigned long long)ohi << 32) | olo;
            if (o > best) best = o;
        }
        if (lane == 0) s_wmax[wave] = best;
        __syncthreads();
        if (wave == 0) {
            unsigned long long v = s_wmax[lane];
            for (int off = 16; off > 0; off >>= 1) {
                unsigned lo = (unsigned)v, hi = (unsigned)(v >> 32);
                unsigned olo = __shfl_xor(lo, off, 32);
                unsigned ohi = __shfl_xor(hi, off, 32);
                unsigned long long o = ((unsigned long long)ohi << 32) | olo;
                if (o > v) v = o;
            }
            if (lane == 0) {
                unsigned j = 0xFFFFFFFFu - (unsigned)(v & 0xFFFFFFFFull);
                s_selidx[k] = j;
                s_scores[j] = -1.0f;   // remove winner
            }
        }
        __syncthreads();
    }

    // ---- gather selected boxes; write sel output; compute keep0 ----
    const float thresh = *thresh_p;
    for (int t = tid; t < KNMS * 6; t += 1024) {
        int k = t / 6, c = t % 6;
        float v = boxes[((size_t)b * NBOX + s_selidx[k]) * 6 + c];
        s_selbox[t] = v;
        out_sel[((size_t)b * KNMS + k) * 6 + c] = v;
    }
    for (int t = tid; t < KNMS; t += 1024) {
        float conf = boxes[((size_t)b * NBOX + s_selidx[t]) * 6 + 5];
        s_keep[t] = (conf > thresh) ? 1 : 0;
    }
    __syncthreads();

    // ---- greedy per-class NMS on 100 boxes: single wave32, zero barriers ----
    if (tid < 32) {
        float x1[4] = {0,0,0,0}, y1[4] = {0,0,0,0}, x2[4] = {0,0,0,0};
        float y2[4] = {0,0,0,0}, cl[4] = {0,0,0,0}, ar[4] = {0,0,0,0};
        unsigned km = 0;                 // keep bits for j = tid + 32*slot
        #pragma unroll
        for (int s = 0; s < 4; ++s) {
            int j = tid + 32 * s;
            if (j < KNMS) {
                x1[s] = s_selbox[j * 6 + 0]; y1[s] = s_selbox[j * 6 + 1];
                x2[s] = s_selbox[j * 6 + 2]; y2[s] = s_selbox[j * 6 + 3];
                cl[s] = s_selbox[j * 6 + 4];
                ar[s] = fmaxf(x2[s] - x1[s], 0.f) * fmaxf(y2[s] - y1[s], 0.f);
                km |= (unsigned)s_keep[j] << s;
            }
        }
        for (int i = 0; i < KNMS; ++i) {
            unsigned mi = __shfl(km, i & 31, 32);      // live keep[i]
            int cur = (mi >> (i >> 5)) & 1;
            float xi1 = s_selbox[i * 6 + 0], yi1 = s_selbox[i * 6 + 1];
            float xi2 = s_selbox[i * 6 + 2], yi2 = s_selbox[i * 6 + 3];
            float cli = s_selbox[i * 6 + 4];
            float ai  = fmaxf(xi2 - xi1, 0.f) * fmaxf(yi2 - yi1, 0.f);
            #pragma unroll
            for (int s = 0; s < 4; ++s) {
                int j = tid + 32 * s;
                if (j < KNMS && j > i) {
                    float xx1 = fmaxf(xi1, x1[s]), yy1 = fmaxf(yi1, y1[s]);
                    float xx2 = fminf(xi2, x2[s]), yy2 = fminf(yi2, y2[s]);
                    float inter = fmaxf(xx2 - xx1, 0.f) * fmaxf(yy2 - yy1, 0.f);
                    float iou = inter / fmaxf(ai + ar[s] - inter, 1e-9f);
                    if (cur && (cli == cl[s]) && (iou > IOU_T)) km &= ~(1u << s);
                }
            }
        }
        #pragma unroll
        for (int s = 0; s < 4; ++s) {
            int j = tid + 32 * s;
            if (j < KNMS)
                out_keep[(size_t)b * KNMS + j] = ((km >> s) & 1) ? 1.0f : 0.0f;
        }
    }
}

// ---------------------------------------------------------------------------
extern "C" void kernel_launch(void* const* d_in, const int* in_sizes, int n_in,
                              void* d_out, int out_size, void* d_ws, size_t ws_size,
                              hipStream_t stream)
{
    const float* out13   = (const float*)d_in[0];
    const float* out26   = (const float*)d_in[1];
    const float* out52   = (const float*)d_in[2];
    const float* anchors = (const float*)d_in[3];   // [3,3,2]
    const float* thresh  = (const float*)d_in[4];

    float* boxes  = (float*)d_ws;                                        // B*NBOX*6 floats
    float* scores = (float*)((char*)d_ws +
                             (size_t)BATCH * NBOX * 6 * sizeof(float));  // B*NPAD floats (16B aligned)

    float* out_sel  = (float*)d_out;                  // [B,100,6]
    float* out_keep = out_sel + BATCH * KNMS * 6;     // [B,100]

    {   // scale 13, stride 32, anchors row 0, boxBase 0
        constexpr int total = BATCH * 3 * 13 * 13;
        decode_kernel<13><<<(total + 255) / 256, 256, 0, stream>>>(
            out13, anchors + 0 * 6, thresh, 32.0f, 0, boxes, scores);
    }
    {   // scale 26, stride 16, anchors row 1, boxBase 507
        constexpr int total = BATCH * 3 * 26 * 26;
        decode_kernel<26><<<(total + 255) / 256, 256, 0, stream>>>(
            out26, anchors + 1 * 6, thresh, 16.0f, 507, boxes, scores);
    }
    {   // scale 52, stride 8, anchors row 2, boxBase 2535
        constexpr int total = BATCH * 3 * 52 * 52;
        decode_kernel<52><<<(total + 255) / 256, 256, 0, stream>>>(
            out52, anchors + 2 * 6, thresh, 8.0f, 2535, boxes, scores);
    }

    topk_nms_kernel<<<BATCH, 1024, 0, stream>>>(boxes, scores, thresh, out_sel, out_keep);
}